// PSROIAlign_4080218931861
// MI455X (gfx1250) — compile-verified
//
#include <hip/hip_runtime.h>
#include <stdint.h>

// Problem constants (match reference)
#define SCALE   0.0625f
#define PP      7            // ROI_SIZE
#define SS      2            // SAMPLING_RATIO
#define DD      5            // POOLED_DIM
#define Hn      128
#define Wn      128
#define Cn      (DD * PP * PP)   // 245 channels == outputs per ROI

// One block per ROI. 256 threads = 8 wave32s.
// Thread tid handles sample s = tid&3 of output o = (tid>>2) + 64*it, it=0..3.
// The 4 lanes of a quad cover the 2x2 sampling grid of one output bin and
// reduce via lane-permute shuffles -> gathers within a quad hit the same
// channel slice and a tiny (y,x) neighborhood (good L0/L2 transaction merge).
__global__ __launch_bounds__(256) void psroi_align_kernel(
    const float* __restrict__ data,   // (B, 245, 128, 128)
    const float* __restrict__ rois,   // (R, 5)
    float* __restrict__ out)          // (R, 245)
{
    __shared__ float s_roi[8];

    const int tid = threadIdx.x;
    const int r   = blockIdx.x;

    // ---- CDNA5 async-to-LDS stage of the 5-float ROI row -------------------
    // Lanes 0..4 of wave 0 each pull one dword straight into LDS (bypasses
    // VGPRs, tracked by ASYNCcnt). Low 32 bits of a generic shared-memory
    // address are the wave-relative LDS byte offset on gfx1250.
    if (tid < 5) {
        uint32_t lds_off = (uint32_t)(uintptr_t)(&s_roi[tid]);
        uint64_t gaddr   = (uint64_t)(uintptr_t)(rois + (size_t)r * 5 + tid);
        asm volatile("global_load_async_to_lds_b32 %0, %1, off"
                     :: "v"(lds_off), "v"(gaddr)
                     : "memory");
    }
    asm volatile("s_wait_asynccnt 0" ::: "memory");
    __syncthreads();

    const int   b  = (int)s_roi[0];
    const float x1 = s_roi[1] * SCALE - 0.5f;
    const float y1 = s_roi[2] * SCALE - 0.5f;
    const float x2 = s_roi[3] * SCALE - 0.5f;
    const float y2 = s_roi[4] * SCALE - 0.5f;

    const float roi_w = fmaxf(x2 - x1, 0.1f);
    const float roi_h = fmaxf(y2 - y1, 0.1f);
    const float bin_w = roi_w * (1.0f / PP);
    const float bin_h = roi_h * (1.0f / PP);

    const float* __restrict__ img = data + (size_t)b * Cn * Hn * Wn;

    const int s  = tid & 3;           // sample id within quad
    const int sy = s >> 1;
    const int sx = s & 1;
    const int o0 = tid >> 2;          // first output channel for this quad

    #pragma unroll
    for (int it = 0; it < 4; ++it) {
        const int o = o0 + it * 64;   // output channel index (== PS channel)
        float val = 0.0f;

        if (o < Cn) {
            const int pw = o % PP;
            const int ph = (o / PP) % PP;

            // Sample coordinates (match reference exactly)
            const float y = y1 + (float)ph * bin_h
                          + ((float)sy + 0.5f) * bin_h * (1.0f / SS);
            const float x = x1 + (float)pw * bin_w
                          + ((float)sx + 0.5f) * bin_w * (1.0f / SS);

            const bool vy = (y >= -1.0f) && (y <= (float)Hn);
            const bool vx = (x >= -1.0f) && (x <= (float)Wn);

            const float cy = fminf(fmaxf(y, 0.0f), (float)(Hn - 1));
            const float cx = fminf(fmaxf(x, 0.0f), (float)(Wn - 1));
            const int   yl = (int)floorf(cy);
            const int   xl = (int)floorf(cx);
            const int   yh = min(yl + 1, Hn - 1);
            const int   xh = min(xl + 1, Wn - 1);
            const float fy = cy - (float)yl;
            const float fx = cx - (float)xl;
            const float gy = 1.0f - fy;
            const float gx = 1.0f - fx;

            // Position-sensitive: channel == o for this output element.
            const float* __restrict__ f = img + (size_t)o * (Hn * Wn);
            const float v11 = f[yl * Wn + xl];
            const float v12 = f[yl * Wn + xh];
            const float v21 = f[yh * Wn + xl];
            const float v22 = f[yh * Wn + xh];

            val = gy * gx * v11 + gy * fx * v12
                + fy * gx * v21 + fy * fx * v22;
            val = (vy && vx) ? val : 0.0f;
        }

        // Quad reduction over the 2x2 samples (lane-permute, no LDS traffic)
        val += __shfl_xor(val, 1, 32);
        val += __shfl_xor(val, 2, 32);

        if (s == 0 && o < Cn) {
            out[(size_t)r * Cn + o] = val * 0.25f;   // mean over S*S samples
        }
    }
}

extern "C" void kernel_launch(void* const* d_in, const int* in_sizes, int n_in,
                              void* d_out, int out_size, void* d_ws, size_t ws_size,
                              hipStream_t stream) {
    const float* data = (const float*)d_in[0];   // (B, 245, 128, 128) f32
    const float* rois = (const float*)d_in[1];   // (R, 5) f32
    float* out        = (float*)d_out;           // (R, 245) f32

    const int R = in_sizes[1] / 5;               // 4000
    psroi_align_kernel<<<R, 256, 0, stream>>>(data, rois, out);
}